// CustomActivation_53480932769916
// MI455X (gfx1250) — compile-verified
//
#include <hip/hip_runtime.h>

// knot_gelu(x) = 0.5*x*(1+tanh(c1*(x + c2*k^3))),  k = ceil(x)*(ceil(x)+1)
// Memory-bound streaming kernel: 1.07 GB traffic, ~46 us floor @ 23.3 TB/s.
// WMMA intentionally unused: no matmul structure in an elementwise op.

typedef float v4f __attribute__((ext_vector_type(4)));

constexpr float kSqrt2OverPi = 0.79788456080286535588f;  // sqrt(2/pi)
constexpr float kGeluCoef    = 0.044715f;
constexpr float kNeg2Log2e   = -2.88539008177792681472f; // -2*log2(e)

__device__ __forceinline__ float knot_gelu(float x) {
  float m  = __builtin_ceilf(x);                       // v_ceil_f32
  float k  = m * (m + 1.0f);                           // knot_invariant
  float k3 = (k * k) * k;
  float z  = kSqrt2OverPi * __builtin_fmaf(kGeluCoef, k3, x);
  // 0.5*x*(1+tanh(z)) == x * sigmoid(2z) == x / (1 + exp(-2z))
  float e  = __builtin_amdgcn_exp2f(kNeg2Log2e * z);   // v_exp_f32
  return x * __builtin_amdgcn_rcpf(1.0f + e);          // v_rcp_f32
}

constexpr int kThreads = 256;  // 8 wave32 per block
constexpr int kItems   = 4;    // four b128 transactions in flight per thread

__global__ __launch_bounds__(kThreads)
void knot_gelu_v4(const v4f* __restrict__ in, v4f* __restrict__ out,
                  unsigned long long n4) {
  const unsigned long long tile =
      (unsigned long long)blockIdx.x * (unsigned long long)(kThreads * kItems);
  const unsigned long long idx0 = tile + threadIdx.x;

  if (tile + (unsigned long long)(kThreads * kItems) <= n4) {
    // Full tile: issue all 4 NT b128 loads before any compute (MLP), then
    // VALU phase, then 4 NT b128 stores.
    v4f v[kItems];
#pragma unroll
    for (int j = 0; j < kItems; ++j)
      v[j] = __builtin_nontemporal_load(in + idx0 + j * kThreads);
#pragma unroll
    for (int j = 0; j < kItems; ++j) {
      v4f r;
      r.x = knot_gelu(v[j].x);
      r.y = knot_gelu(v[j].y);
      r.z = knot_gelu(v[j].z);
      r.w = knot_gelu(v[j].w);
      v[j] = r;
    }
#pragma unroll
    for (int j = 0; j < kItems; ++j)
      __builtin_nontemporal_store(v[j], out + idx0 + j * kThreads);
  } else {
    // Guarded tail tile.
#pragma unroll
    for (int j = 0; j < kItems; ++j) {
      const unsigned long long i = idx0 + (unsigned long long)j * kThreads;
      if (i < n4) {
        v4f v = __builtin_nontemporal_load(in + i);
        v4f r;
        r.x = knot_gelu(v.x);
        r.y = knot_gelu(v.y);
        r.z = knot_gelu(v.z);
        r.w = knot_gelu(v.w);
        __builtin_nontemporal_store(r, out + i);
      }
    }
  }
}

// Scalar tail for n % 4 != 0 (not hit for this shape, kept for generality).
__global__ void knot_gelu_tail(const float* __restrict__ in,
                               float* __restrict__ out,
                               unsigned long long start, unsigned long long n) {
  unsigned long long i =
      start + (unsigned long long)blockIdx.x * blockDim.x + threadIdx.x;
  if (i < n) out[i] = knot_gelu(in[i]);
}

extern "C" void kernel_launch(void* const* d_in, const int* in_sizes, int n_in,
                              void* d_out, int out_size, void* d_ws,
                              size_t ws_size, hipStream_t stream) {
  const float* x = (const float*)d_in[0];
  float* y = (float*)d_out;

  const unsigned long long n  = (unsigned long long)in_sizes[0];
  const unsigned long long n4 = n / 4;
  const unsigned long long per_block =
      (unsigned long long)kThreads * (unsigned long long)kItems;
  const unsigned long long blocks = (n4 + per_block - 1) / per_block;

  if (blocks) {
    knot_gelu_v4<<<(unsigned)blocks, kThreads, 0, stream>>>(
        (const v4f*)x, (v4f*)y, n4);
  }
  const unsigned long long rem = n - n4 * 4ull;
  if (rem) {
    knot_gelu_tail<<<1, 64, 0, stream>>>(x, y, n4 * 4ull, n);
  }
}